// DCCFEncoder_74019466379644
// MI455X (gfx1250) — compile-verified
//
#include <hip/hip_runtime.h>
#include <hip/hip_bf16.h>

typedef __attribute__((ext_vector_type(2))) float v2f;
typedef __attribute__((ext_vector_type(8))) float v8f;

#define DD 64   // embedding dim
#define KK 128  // intent dim

// ---------------- elementwise helpers ----------------

__global__ void zero_kernel(float* __restrict__ p, long n) {
  long i = (long)blockIdx.x * blockDim.x + threadIdx.x;
  if (i < n) p[i] = 0.f;
}

__global__ void init_x_kernel(const float* __restrict__ ue, const float* __restrict__ ie,
                              float* __restrict__ x, float* __restrict__ sum,
                              long UD, long ND) {
  long i = (long)blockIdx.x * blockDim.x + threadIdx.x;
  if (i < ND) {
    float v = (i < UD) ? ue[i] : ie[i - UD];
    x[i] = v;
    sum[i] = v;
  }
}

__global__ void combine_kernel(const float* __restrict__ x, const float* __restrict__ gnn,
                               const float* __restrict__ intn, const float* __restrict__ gaa,
                               const float* __restrict__ iaa, float* __restrict__ xnext,
                               float* __restrict__ sum, long ND) {
  long i = (long)blockIdx.x * blockDim.x + threadIdx.x;
  if (i < ND) {
    float v = gnn[i] + intn[i] + gaa[i] + iaa[i] + x[i];
    xnext[i] = v;
    sum[i] += v;
  }
}

__global__ void mean_kernel(const float* __restrict__ sum, float* __restrict__ out, long ND) {
  long i = (long)blockIdx.x * blockDim.x + threadIdx.x;
  if (i < ND) out[i] = sum[i] * (1.f / 3.f);
}

// ---------------- edge kernels (L2-resident gathers + f32 atomics) ----------------
// 16 threads per edge, each thread owns 4 contiguous features (float4 gather).

__global__ void spmm_edges_kernel(const float* __restrict__ vals, const int* __restrict__ h,
                                  const int* __restrict__ t, const float* __restrict__ x,
                                  float* __restrict__ out, long E) {
  long tid = (long)blockIdx.x * blockDim.x + threadIdx.x;
  long e = tid >> 4;
  if (e >= E) return;
  int d = (int)(tid & 15) * 4;
  if (d == 0 && e + 2048 < E)
    __builtin_prefetch(x + (long)t[e + 2048] * DD, 0, 1);  // global_prefetch_b8
  int hi = h[e], ti = t[e];
  float v = vals[e];
  const float4 xv = *(const float4*)(x + (long)ti * DD + d);
  float* o = out + (long)hi * DD + d;
  atomicAdd(o + 0, v * xv.x);
  atomicAdd(o + 1, v * xv.y);
  atomicAdd(o + 2, v * xv.z);
  atomicAdd(o + 3, v * xv.w);
}

// per-node 1/max(||a_n||, 1e-12): 16 lanes per node
__global__ void inv_norm_kernel(const float* __restrict__ a, float* __restrict__ invn, long N) {
  long tid = (long)blockIdx.x * blockDim.x + threadIdx.x;
  long n = tid >> 4;
  if (n >= N) return;
  int d = (int)(tid & 15) * 4;
  const float4 v = *(const float4*)(a + n * DD + d);
  float s = v.x * v.x + v.y * v.y + v.z * v.z + v.w * v.w;
  for (int off = 1; off < 16; off <<= 1) s += __shfl_xor(s, off, 32);
  if ((tid & 15) == 0) invn[n] = 1.f / fmaxf(sqrtf(s), 1e-12f);
}

// alpha_e = (cos(a[h],a[t]) + 1)/2 ; rowsum[h] += alpha
__global__ void edge_alpha_kernel(const float* __restrict__ a, const float* __restrict__ invn,
                                  const int* __restrict__ h, const int* __restrict__ t,
                                  float* __restrict__ alpha, float* __restrict__ rowsum, long E) {
  long tid = (long)blockIdx.x * blockDim.x + threadIdx.x;
  long e = tid >> 4;
  if (e >= E) return;
  int d = (int)(tid & 15) * 4;
  int hi = h[e], ti = t[e];
  const float4 hv = *(const float4*)(a + (long)hi * DD + d);
  const float4 tv = *(const float4*)(a + (long)ti * DD + d);
  float s = hv.x * tv.x + hv.y * tv.y + hv.z * tv.z + hv.w * tv.w;
  for (int off = 1; off < 16; off <<= 1) s += __shfl_xor(s, off, 32);
  if ((tid & 15) == 0) {
    float al = 0.5f * (s * invn[hi] * invn[ti] + 1.f);
    alpha[e] = al;
    atomicAdd(rowsum + hi, al);
  }
}

// out[h] += (alpha_e / rowsum[h]) * x[t]
__global__ void spmm_adaptive_kernel(const float* __restrict__ alpha,
                                     const float* __restrict__ rowsum,
                                     const int* __restrict__ h, const int* __restrict__ t,
                                     const float* __restrict__ x, float* __restrict__ out,
                                     long E) {
  long tid = (long)blockIdx.x * blockDim.x + threadIdx.x;
  long e = tid >> 4;
  if (e >= E) return;
  int d = (int)(tid & 15) * 4;
  if (d == 0 && e + 2048 < E)
    __builtin_prefetch(x + (long)t[e + 2048] * DD, 0, 1);
  int hi = h[e], ti = t[e];
  float rs = rowsum[hi];
  float w = (rs > 0.f) ? (alpha[e] / rs) : 0.f;
  const float4 xv = *(const float4*)(x + (long)ti * DD + d);
  float* o = out + (long)hi * DD + d;
  atomicAdd(o + 0, w * xv.x);
  atomicAdd(o + 1, w * xv.y);
  atomicAdd(o + 2, w * xv.z);
  atomicAdd(o + 3, w * xv.w);
}

// ---------------- fused intent kernel (WMMA fp32) ----------------
// Per wave: one 16-row tile.
//   S  = A(16x64) @ W(64x128)         via v_wmma_f32_16x16x4_f32
//   P  = softmax_row(S)               in-register, 16-lane shfl reductions
//   O  = P(16x128) @ W^T(128x64)      via v_wmma_f32_16x16x4_f32 (A-frags re-laid through LDS)
__global__ __launch_bounds__(256) void intent_kernel(const float* __restrict__ x,
                                                     const float* __restrict__ Wu,
                                                     const float* __restrict__ Wi,
                                                     float* __restrict__ intn,
                                                     long U, long N) {
  __shared__ float Plds[8 * 16 * KK];  // 64 KB: per-wave P staging
  const int wave = threadIdx.x >> 5;
  const int lane = threadIdx.x & 31;
  const int half = lane >> 4;  // 0: K/N rows 0..1 | 1: rows 2..3 of each frag
  const int l16 = lane & 15;
  long nTiles = N >> 4;
  long tile = (long)blockIdx.x * 8 + wave;
  if (tile >= nTiles) tile = nTiles - 1;  // wave-uniform clamp; duplicate stores are identical
  const float* W = (tile * 16 < U) ? Wu : Wi;  // U,I multiples of 16: no straddling
  const float* A = x + tile * 16 * DD;

  // ---- GEMM1: accumulate 8 16x16 f32 tiles over K=64 in steps of 4 ----
  v8f acc[8] = {};
  for (int k0 = 0; k0 < DD; k0 += 4) {
    int ka = k0 + half * 2;
    v2f a;
    a[0] = A[(long)l16 * DD + ka];
    a[1] = A[(long)l16 * DD + ka + 1];
#pragma unroll
    for (int nb = 0; nb < 8; ++nb) {
      int n = nb * 16 + l16;
      v2f b;
      b[0] = W[ka * KK + n];
      b[1] = W[(ka + 1) * KK + n];
      acc[nb] = __builtin_amdgcn_wmma_f32_16x16x4_f32(false, a, false, b, (short)0, acc[nb],
                                                      false, false);
    }
  }

  // ---- softmax over the 128 logits of each row ----
  // row M = half*8 + r lives in component r across lanes [half*16, half*16+15]
#pragma unroll
  for (int r = 0; r < 8; ++r) {
    float mm = acc[0][r];
#pragma unroll
    for (int nb = 1; nb < 8; ++nb) mm = fmaxf(mm, acc[nb][r]);
    for (int off = 1; off < 16; off <<= 1) mm = fmaxf(mm, __shfl_xor(mm, off, 32));
    float ss = 0.f;
#pragma unroll
    for (int nb = 0; nb < 8; ++nb) {
      float ev = __expf(acc[nb][r] - mm);
      acc[nb][r] = ev;
      ss += ev;
    }
    for (int off = 1; off < 16; off <<= 1) ss += __shfl_xor(ss, off, 32);
    float inv = 1.f / ss;
#pragma unroll
    for (int nb = 0; nb < 8; ++nb) acc[nb][r] *= inv;
  }

  // ---- stage P to LDS (C-layout -> row-major) for A-fragment reload ----
  float* P = Plds + wave * (16 * KK);
#pragma unroll
  for (int nb = 0; nb < 8; ++nb)
#pragma unroll
    for (int r = 0; r < 8; ++r)
      P[(half * 8 + r) * KK + nb * 16 + l16] = acc[nb][r];
  asm volatile("s_wait_dscnt 0x0" ::: "memory");  // same-wave LDS RAW ordering

  // ---- GEMM2: O = P(16x128) @ W^T(128x64) ----
  v8f acc2[4] = {};
  for (int k0 = 0; k0 < KK; k0 += 4) {
    int ka = k0 + half * 2;
    v2f a;
    a[0] = P[l16 * KK + ka];
    a[1] = P[l16 * KK + ka + 1];
#pragma unroll
    for (int nb = 0; nb < 4; ++nb) {
      int n = nb * 16 + l16;
      v2f b;
      b[0] = W[n * KK + ka];      // B[k][n] = W^T[k][n] = W[n][k]
      b[1] = W[n * KK + ka + 1];
      acc2[nb] = __builtin_amdgcn_wmma_f32_16x16x4_f32(false, a, false, b, (short)0, acc2[nb],
                                                       false, false);
    }
  }

  float* O = intn + tile * 16 * DD;
#pragma unroll
  for (int nb = 0; nb < 4; ++nb)
#pragma unroll
    for (int r = 0; r < 8; ++r)
      O[(long)(half * 8 + r) * DD + nb * 16 + l16] = acc2[nb][r];
}

// ---------------- host-side launch ----------------

extern "C" void kernel_launch(void* const* d_in, const int* in_sizes, int n_in,
                              void* d_out, int out_size, void* d_ws, size_t ws_size,
                              hipStream_t stream) {
  const float* user_emb = (const float*)d_in[0];
  const float* item_emb = (const float*)d_in[1];
  const float* Wu = (const float*)d_in[2];
  const float* Wi = (const float*)d_in[3];
  const float* adj_vals = (const float*)d_in[4];
  const int* h_idx = (const int*)d_in[5];
  const int* t_idx = (const int*)d_in[6];

  const long U = in_sizes[0] / DD;
  const long I = in_sizes[1] / DD;
  const long N = U + I;
  const long E = in_sizes[4];
  const long UD = U * DD;
  const long ND = N * DD;

  float* out = (float*)d_out;
  float* ws = (float*)d_ws;
  float* x_cur = ws;
  float* x_nxt = ws + ND;
  float* sumb = ws + 2 * ND;
  float* invn = ws + 3 * ND;
  float* alpha = invn + N;
  float* rowsum = alpha + E;

  const int T = 256;
  const int bND = (int)((ND + T - 1) / T);
  const int bEdge = (int)((E * 16 + T - 1) / T);
  const int bNode = (int)((N * 16 + T - 1) / T);
  const long nTiles = N >> 4;
  const int bIntent = (int)((nTiles + 7) / 8);

  init_x_kernel<<<bND, T, 0, stream>>>(user_emb, item_emb, x_cur, sumb, UD, ND);

  for (int k = 0; k < 2; ++k) {
    float* gnn_k = out + ND + (long)k * ND;       // gnn_l slot
    float* int_k = out + 3 * ND + (long)k * ND;   // int_l slot
    float* gaa_k = out + 5 * ND + (long)k * ND;   // gaa_l slot
    float* iaa_k = out + 7 * ND + (long)k * ND;   // iaa_l slot

    // gnn = spmm(adj_vals, x)
    zero_kernel<<<bND, T, 0, stream>>>(gnn_k, ND);
    spmm_edges_kernel<<<bEdge, T, 0, stream>>>(adj_vals, h_idx, t_idx, x_cur, gnn_k, E);

    // intn = softmax(x @ W) @ W^T  (WMMA)
    intent_kernel<<<bIntent, T, 0, stream>>>(x_cur, Wu, Wi, int_k, U, N);

    // gaa: adaptive aggregation on gnn
    inv_norm_kernel<<<bNode, T, 0, stream>>>(gnn_k, invn, N);
    zero_kernel<<<(int)((N + T - 1) / T), T, 0, stream>>>(rowsum, N);
    edge_alpha_kernel<<<bEdge, T, 0, stream>>>(gnn_k, invn, h_idx, t_idx, alpha, rowsum, E);
    zero_kernel<<<bND, T, 0, stream>>>(gaa_k, ND);
    spmm_adaptive_kernel<<<bEdge, T, 0, stream>>>(alpha, rowsum, h_idx, t_idx, x_cur, gaa_k, E);

    // iaa: adaptive aggregation on intn
    inv_norm_kernel<<<bNode, T, 0, stream>>>(int_k, invn, N);
    zero_kernel<<<(int)((N + T - 1) / T), T, 0, stream>>>(rowsum, N);
    edge_alpha_kernel<<<bEdge, T, 0, stream>>>(int_k, invn, h_idx, t_idx, alpha, rowsum, E);
    zero_kernel<<<bND, T, 0, stream>>>(iaa_k, ND);
    spmm_adaptive_kernel<<<bEdge, T, 0, stream>>>(alpha, rowsum, h_idx, t_idx, x_cur, iaa_k, E);

    // x_{k+1} = gnn + intn + gaa + iaa + x ; sum += x_{k+1}
    combine_kernel<<<bND, T, 0, stream>>>(x_cur, gnn_k, int_k, gaa_k, iaa_k, x_nxt, sumb, ND);
    float* tmp = x_cur; x_cur = x_nxt; x_nxt = tmp;
  }

  // ua/ia = mean of the 3 embedding stages
  mean_kernel<<<bND, T, 0, stream>>>(sumb, out, ND);
}